// FD_9904194585032
// MI455X (gfx1250) — compile-verified
//
#include <hip/hip_runtime.h>

typedef float v2f __attribute__((ext_vector_type(2)));
typedef float v4f __attribute__((ext_vector_type(4)));
typedef float v8f __attribute__((ext_vector_type(8)));

#define NB   8
#define CC   256
#define HH   128
#define WW   128
#define HW   (HH * WW)          // 16384
#define PLANES (NB * CC)        // 2048
#define NOUT 72                 // GROUP * K * K
#define CG   32                 // channels per group

// ---------------------------------------------------------------------------
// Kernel 1: global average pool over H*W for each (n,c) plane.
// One block per plane, 256 threads, float4 loads, shuffle+LDS reduction.
// ---------------------------------------------------------------------------
__global__ __launch_bounds__(256) void pool_kernel(const float* __restrict__ x,
                                                   float* __restrict__ pooled) {
    const int plane = blockIdx.x;                      // n*256 + c
    const v4f* p = (const v4f*)(x + (size_t)plane * HW);
    float s = 0.f;
    #pragma unroll 4
    for (int i = threadIdx.x; i < HW / 4; i += 256) {
        v4f v = p[i];
        s += (v.x + v.y) + (v.z + v.w);
    }
    // wave32 reduction
    #pragma unroll
    for (int off = 16; off > 0; off >>= 1)
        s += __shfl_down(s, off, 32);
    __shared__ float red[8];
    const int wave = threadIdx.x >> 5;
    if ((threadIdx.x & 31) == 0) red[wave] = s;
    __syncthreads();
    if (threadIdx.x == 0) {
        float t = 0.f;
        #pragma unroll
        for (int i = 0; i < 8; ++i) t += red[i];
        pooled[plane] = t * (1.0f / (float)HW);
    }
}

// ---------------------------------------------------------------------------
// Kernel 2: lf = pooled(8x256) @ conv_w^T(256x72) via V_WMMA_F32_16X16X4_F32,
// then BN + softmax(9) -> filt[8][8][9]. Single wave32 workgroup.
//   M = 16 (8 batches, zero padded), N = 5 tiles of 16 (72 padded to 80),
//   K = 256 in 64 steps of 4.
// ---------------------------------------------------------------------------
__global__ __launch_bounds__(32) void filt_kernel(const float* __restrict__ pooled,   // [8][256]
                                                  const float* __restrict__ conv_w,   // [72][256]
                                                  const float* __restrict__ bn_gamma,
                                                  const float* __restrict__ bn_beta,
                                                  const float* __restrict__ bn_mean,
                                                  const float* __restrict__ bn_var,
                                                  float* __restrict__ filt) {         // [8*8*9]
    __shared__ float lf[NB][80];
    const int lane  = threadIdx.x;           // 0..31
    const int m     = lane & 15;             // row within 16-high fragment
    const int khalf = (lane >> 4) * 2;       // 0 for lanes 0-15, 2 for lanes 16-31

    v8f acc[5];
    #pragma unroll
    for (int t = 0; t < 5; ++t) acc[t] = (v8f){0.f,0.f,0.f,0.f,0.f,0.f,0.f,0.f};

    for (int k = 0; k < CC; k += 4) {
        // A fragment: 16x4 f32. Lane (m, half): VGPR0=K(khalf), VGPR1=K(khalf+1)
        v2f a = *(const v2f*)(pooled + (m & 7) * CC + k + khalf);
        if (m >= NB) { a.x = 0.f; a.y = 0.f; }
        #pragma unroll
        for (int t = 0; t < 5; ++t) {
            // B fragment: 4x16 f32. Lane holds column N = t*16 + m, rows khalf..khalf+1
            const int o  = t * 16 + m;
            const int oc = (o < NOUT) ? o : (NOUT - 1);
            v2f b = *(const v2f*)(conv_w + oc * CC + k + khalf);
            if (o >= NOUT) { b.x = 0.f; b.y = 0.f; }
            acc[t] = __builtin_amdgcn_wmma_f32_16x16x4_f32(
                false, a, false, b, (short)0, acc[t], false, false);
        }
    }

    // D layout: VGPR v, lanes 0-15 -> (M=v, N=lane). We only need M=0..7.
    if (lane < 16) {
        #pragma unroll
        for (int t = 0; t < 5; ++t)
            #pragma unroll
            for (int v = 0; v < 8; ++v)
                lf[v][t * 16 + lane] = acc[t][v];
    }
    __syncthreads();

    // BN + softmax over 9 taps for each of the 64 (n, group) pairs.
    for (int p = lane; p < NB * 8; p += 32) {
        const int n = p >> 3;
        const int g = p & 7;
        float v[9];
        float mx = -3.402823e38f;
        #pragma unroll
        for (int j = 0; j < 9; ++j) {
            const int o = g * 9 + j;
            float t = (lf[n][o] - bn_mean[o]) * rsqrtf(bn_var[o] + 1e-5f) * bn_gamma[o] + bn_beta[o];
            v[j] = t;
            mx = fmaxf(mx, t);
        }
        float s = 0.f;
        #pragma unroll
        for (int j = 0; j < 9; ++j) { v[j] = __expf(v[j] - mx); s += v[j]; }
        const float inv = 1.0f / s;
        #pragma unroll
        for (int j = 0; j < 9; ++j) filt[p * 9 + j] = v[j] * inv;
    }
}

// ---------------------------------------------------------------------------
// Kernel 3: 3x3 reflect-padded weighted smoothing with per-(n,group) weights.
// Grid: 2048 planes; block (32,4): each wave owns a 32-row strip, each lane
// owns 4 columns (float4). Rows are loaded once and slid vertically:
//   low(h) = s0(h-1) + s1(h) + s2(h+1),  s_k(r) = weight-row-k horiz combo.
// ---------------------------------------------------------------------------
struct RowT { v4f s0, s1, s2, m; };

__device__ __forceinline__ v4f hcombo(float l, v4f m, float r,
                                      float wl, float wm, float wr) {
    v4f s;
    s.x = fmaf(wl, l,   fmaf(wm, m.x, wr * m.y));
    s.y = fmaf(wl, m.x, fmaf(wm, m.y, wr * m.z));
    s.z = fmaf(wl, m.y, fmaf(wm, m.z, wr * m.w));
    s.w = fmaf(wl, m.z, fmaf(wm, m.w, wr * r));
    return s;
}

__device__ __forceinline__ int rref(int r) {
    return r < 0 ? -r : (r > HH - 1 ? 2 * (HH - 1) - r : r);
}

__device__ __forceinline__ RowT load_row(const float* __restrict__ xp, int r, int t,
                                         float w00, float w01, float w02,
                                         float w10, float w11, float w12,
                                         float w20, float w21, float w22) {
    const float* row = xp + r * WW;
    RowT o;
    o.m = *(const v4f*)(row + 4 * t);
    // column reflect: col -1 -> 1 (== m.y), col 128 -> 126 (== m.z)
    const float left  = (t == 0)      ? o.m.y : row[4 * t - 1];
    const float right = (t == 31)     ? o.m.z : row[4 * t + 4];
    o.s0 = hcombo(left, o.m, right, w00, w01, w02);
    o.s1 = hcombo(left, o.m, right, w10, w11, w12);
    o.s2 = hcombo(left, o.m, right, w20, w21, w22);
    return o;
}

__global__ __launch_bounds__(128) void smooth_kernel(const float* __restrict__ x,
                                                     const float* __restrict__ filt,
                                                     float* __restrict__ low,
                                                     float* __restrict__ high) {
    const int plane = blockIdx.x;            // n*256 + c
    const int n = plane >> 8;
    const int g = (plane & 255) >> 5;        // c / CG
    const int t = threadIdx.x;               // lane -> columns 4t..4t+3
    const int h0 = threadIdx.y * 32;         // row strip

    const float* fw = filt + (n * 8 + g) * 9;
    const float w00 = fw[0], w01 = fw[1], w02 = fw[2];
    const float w10 = fw[3], w11 = fw[4], w12 = fw[5];
    const float w20 = fw[6], w21 = fw[7], w22 = fw[8];

    const float* __restrict__ xp = x + (size_t)plane * HW;
    float* __restrict__ lo = low  + (size_t)plane * HW;
    float* __restrict__ hi = high + (size_t)plane * HW;

    // prime the pipeline: s0 of row h0-1 (reflected) and full triple of row h0
    v4f prev_s0 = load_row(xp, rref(h0 - 1), t,
                           w00, w01, w02, w10, w11, w12, w20, w21, w22).s0;
    RowT cur = load_row(xp, rref(h0), t,
                        w00, w01, w02, w10, w11, w12, w20, w21, w22);

    #pragma unroll 2
    for (int h = h0; h < h0 + 32; ++h) {
        const int rn = rref(h + 1);
        if (h + 2 < HH)
            __builtin_prefetch(xp + (h + 2) * WW + 4 * t, 0, 0);
        RowT nxt = load_row(xp, rn, t,
                            w00, w01, w02, w10, w11, w12, w20, w21, w22);
        v4f lo4;
        lo4.x = prev_s0.x + cur.s1.x + nxt.s2.x;
        lo4.y = prev_s0.y + cur.s1.y + nxt.s2.y;
        lo4.z = prev_s0.z + cur.s1.z + nxt.s2.z;
        lo4.w = prev_s0.w + cur.s1.w + nxt.s2.w;
        v4f hi4;
        hi4.x = cur.m.x - lo4.x;
        hi4.y = cur.m.y - lo4.y;
        hi4.z = cur.m.z - lo4.z;
        hi4.w = cur.m.w - lo4.w;
        __builtin_nontemporal_store(lo4, (v4f*)(lo + h * WW + 4 * t));
        __builtin_nontemporal_store(hi4, (v4f*)(hi + h * WW + 4 * t));
        prev_s0 = cur.s0;
        cur = nxt;
    }
}

// ---------------------------------------------------------------------------
extern "C" void kernel_launch(void* const* d_in, const int* in_sizes, int n_in,
                              void* d_out, int out_size, void* d_ws, size_t ws_size,
                              hipStream_t stream) {
    const float* x        = (const float*)d_in[0];
    const float* conv_w   = (const float*)d_in[1];
    const float* bn_gamma = (const float*)d_in[2];
    const float* bn_beta  = (const float*)d_in[3];
    const float* bn_mean  = (const float*)d_in[4];
    const float* bn_var   = (const float*)d_in[5];

    float* low  = (float*)d_out;
    float* high = low + (size_t)PLANES * HW;

    float* pooled = (float*)d_ws;      // 2048 floats
    float* filt   = pooled + PLANES;   // 576 floats

    pool_kernel<<<PLANES, 256, 0, stream>>>(x, pooled);
    filt_kernel<<<1, 32, 0, stream>>>(pooled, conv_w, bn_gamma, bn_beta,
                                      bn_mean, bn_var, filt);
    smooth_kernel<<<PLANES, dim3(32, 4), 0, stream>>>(x, filt, low, high);
}